// HybridConvNet_65481071409128
// MI455X (gfx1250) — compile-verified
//
#include <hip/hip_runtime.h>
#include <hip/hip_bf16.h>

// MI455X / gfx1250, wave32. Memory-bound: 51.4 MB in -> ~2.2 us HBM floor.
// Conv patch dots via V_WMMA_F32_16X16X4_F32 (A 16x4 layout == 2x2 patch layout).

typedef float v2f __attribute__((ext_vector_type(2)));
typedef float v8f __attribute__((ext_vector_type(8)));

#define NPATCH 196
#define IMG_ELEMS 784

static __device__ __forceinline__ float fast_rcp(float x) {
#if __has_builtin(__builtin_amdgcn_rcpf)
    return __builtin_amdgcn_rcpf(x);
#else
    return 1.0f / x;
#endif
}

static __device__ __forceinline__ float fast_sigmoid(float x) {
    return fast_rcp(1.0f + __expf(-x));
}

// ---------------------------------------------------------------------------
// Setup: compose the linear chains into tiny matrices (runs once per launch).
//   ws[0..15]  M4 : qfc_w(4,784) @ fc_map_w(784,4)
//   ws[16..19] b4 : qfc_w @ fc_map_b + qfc_b
//   ws[20..35] Wc : qcnn_w1(4,8) @ to_qcnn_w(8,4)
//   ws[36..39] bc : qcnn_w1 @ to_qcnn_b + qcnn_b1
//   ws[40..43] w2 : qcnn_w2
//   ws[44]     b2 : qcnn_b2
// ---------------------------------------------------------------------------
__global__ void __launch_bounds__(256)
setup_compose(const float* __restrict__ fc_map_w, const float* __restrict__ fc_map_b,
              const float* __restrict__ qfc_w,    const float* __restrict__ qfc_b,
              const float* __restrict__ to_qcnn_w,const float* __restrict__ to_qcnn_b,
              const float* __restrict__ qcnn_w1,  const float* __restrict__ qcnn_b1,
              const float* __restrict__ qcnn_w2,  const float* __restrict__ qcnn_b2,
              float* __restrict__ ws)
{
    __shared__ float part[16][17];
    const int t = threadIdx.x;           // 256 threads
    const int e = t & 15;                // output entry
    const int sl = t >> 4;               // K-slice (16 slices of 49)

    {   // M4: 16 entries x 16 K-slices
        const int q = e >> 2, i = e & 3;
        float s = 0.0f;
        const int k0 = sl * 49;
        for (int k = k0; k < k0 + 49; ++k)
            s += qfc_w[q * 784 + k] * fc_map_w[k * 4 + i];
        part[e][sl] = s;
    }
    __syncthreads();
    if (t < 16) {
        float s = 0.0f;
        for (int z = 0; z < 16; ++z) s += part[t][z];
        ws[t] = s;
    }
    __syncthreads();
    if (t < 64) {       // b4 partials: 4 entries x 16 K-slices
        const int q = t >> 4, z = t & 15;
        float s = 0.0f;
        const int k0 = z * 49;
        for (int k = k0; k < k0 + 49; ++k)
            s += qfc_w[q * 784 + k] * fc_map_b[k];
        part[q][z] = s;
    }
    __syncthreads();
    if (t < 4) {
        float s = qfc_b[t];
        for (int z = 0; z < 16; ++z) s += part[t][z];
        ws[16 + t] = s;
    }
    if (t >= 32 && t < 48) {     // Wc (K=8)
        const int e2 = t - 32, hh = e2 >> 2, q = e2 & 3;
        float s = 0.0f;
        for (int mm = 0; mm < 8; ++mm)
            s += qcnn_w1[hh * 8 + mm] * to_qcnn_w[mm * 4 + q];
        ws[20 + e2] = s;
    }
    if (t >= 48 && t < 52) {     // bc
        const int hh = t - 48;
        float s = qcnn_b1[hh];
        for (int mm = 0; mm < 8; ++mm)
            s += qcnn_w1[hh * 8 + mm] * to_qcnn_b[mm];
        ws[36 + hh] = s;
    }
    if (t >= 52 && t < 56) ws[40 + (t - 52)] = qcnn_w2[t - 52];
    if (t == 56)           ws[44] = qcnn_b2[0];
}

// ---------------------------------------------------------------------------
// Main kernel: one wave32 per image.
//   A 16x4 f32 layout: lanes 0-15 hold K=0,1 (patch top-row float2),
//   lanes 16-31 hold K=2,3 (patch bottom-row float2) -> one b64 load fills A.
//   B[k][n] = conv_kernel[k] broadcast. 13 WMMAs cover 196 patches.
//   D row extraction is a branchless one-hot FMA dot (masks loop-invariant).
// ---------------------------------------------------------------------------
__global__ void __launch_bounds__(256)
hybrid_conv_attn(const float* __restrict__ x,
                 const float* __restrict__ kern,
                 const float* __restrict__ cbias,
                 const float* __restrict__ rot,
                 const float* __restrict__ ent,
                 const float* __restrict__ ws,
                 float* __restrict__ out, int B)
{
    const int lane = threadIdx.x & 31;
    const int img  = blockIdx.x * (blockDim.x >> 5) + (threadIdx.x >> 5);
    if (img >= B) return;                      // wave-uniform: EXEC stays all-ones

    const float* __restrict__ im = x + (size_t)img * IMG_ELEMS;

    const int m    = lane & 15;                // A-matrix row M within chunk
    const int half = lane >> 4;                // 0: K=0,1   1: K=2,3
    const int rsel = lane & 7;                 // D row this lane extracts

    // Uniform scalar loads of the 2x2 kernel; divergent-half select via cndmask.
    const float k0 = kern[0], k1 = kern[1], k2 = kern[2], k3 = kern[3];
    v2f bm;
    bm.x = half ? k2 : k0;
    bm.y = half ? k3 : k1;
    const float cb = cbias[0];

    // Issue all 13 A-tile loads up front (software pipelining).
    v2f areg[13];
#pragma unroll
    for (int chunk = 0; chunk < 13; ++chunk) {
        const int p  = chunk * 16 + m;
        const int pc = (p < NPATCH) ? p : 0;   // only chunk 12 can clamp
        const int i  = pc / 14;
        const int j  = pc - i * 14;
        const float2 d = *reinterpret_cast<const float2*>(im + i * 56 + half * 28 + j * 2);
        const float va = (p < NPATCH) ? 1.0f : 0.0f;
        areg[chunk].x = d.x * va;
        areg[chunk].y = d.y * va;
    }

    // Loop-invariant one-hot masks for branchless D-row extraction.
    float msk[8];
#pragma unroll
    for (int k = 0; k < 8; ++k) msk[k] = (rsel == k) ? 1.0f : 0.0f;

    float acc = 0.0f;
#pragma unroll
    for (int chunk = 0; chunk < 13; ++chunk) {
        v8f c = {0.f, 0.f, 0.f, 0.f, 0.f, 0.f, 0.f, 0.f};
        c = __builtin_amdgcn_wmma_f32_16x16x4_f32(false, areg[chunk], false, bm,
                                                  (short)0, c, false, false);
        // s = c[rsel] as an 8-term FMA dot against the one-hot mask.
        float s = c[0] * msk[0];
#pragma unroll
        for (int k = 1; k < 8; ++k) s = fmaf(c[k], msk[k], s);

        // Validity: chunks 0..11 always full; chunk 12 masks rows >= 196.
        float vm = 1.0f;
        if (chunk == 12) {
            const int pm = chunk * 16 + half * 8 + rsel;
            vm = (pm < NPATCH) ? 1.0f : 0.0f;
        }
        acc += vm * fast_sigmoid(s + cb);
    }

    // Wave reduction; each patch counted exactly twice across the wave.
    for (int off = 16; off > 0; off >>= 1) acc += __shfl_xor(acc, off, 32);
    const float v = acc * (1.0f / (2.0f * (float)NPATCH));

    // ---- self-attention over 4 identical features (scalar math) ----
    const float v2s = v * v;
    const float r4 = rot[4], r5 = rot[5], r6 = rot[6], r7 = rot[7];
    const float vr0 = v * rot[8], vr1 = v * rot[9], vr2 = v * rot[10], vr3 = v * rot[11];
    float a4[4];
#pragma unroll
    for (int i4 = 0; i4 < 4; ++i4) {
        const float gi = 0.5f * v2s * rot[i4];
        const float l0 = gi * r4, l1 = gi * r5, l2 = gi * r6, l3 = gi * r7;
        const float mx = fmaxf(fmaxf(l0, l1), fmaxf(l2, l3));
        const float e0x = __expf(l0 - mx), e1x = __expf(l1 - mx);
        const float e2x = __expf(l2 - mx), e3x = __expf(l3 - mx);
        const float den = e0x + e1x + e2x + e3x;
        const float num = e0x * vr0 + e1x * vr1 + e2x * vr2 + e3x * vr3;
        a4[i4] = num * fast_rcp(den);
    }
    const float e0 = ent[0], e1 = ent[1], e2 = ent[2];
    float b4v[4];
#pragma unroll
    for (int i4 = 0; i4 < 4; ++i4)
        b4v[i4] = a4[i4] + e0 * a4[(i4 + 3) & 3]
                         + e1 * a4[(i4 + 2) & 3]
                         + e2 * a4[(i4 + 1) & 3];

    // ---- composed tail: 4x4 -> tanh -> 4x4 -> tanh -> dot4 -> sigmoid ----
    const float* M4  = ws;
    const float* bb4 = ws + 16;
    const float* Wc  = ws + 20;
    const float* bcv = ws + 36;
    const float* w2  = ws + 40;
    const float  b2  = ws[44];

    float tq[4];
#pragma unroll
    for (int q = 0; q < 4; ++q)
        tq[q] = tanhf(M4[q * 4 + 0] * b4v[0] + M4[q * 4 + 1] * b4v[1]
                    + M4[q * 4 + 2] * b4v[2] + M4[q * 4 + 3] * b4v[3] + bb4[q]);
    float hv[4];
#pragma unroll
    for (int hh = 0; hh < 4; ++hh)
        hv[hh] = tanhf(Wc[hh * 4 + 0] * tq[0] + Wc[hh * 4 + 1] * tq[1]
                     + Wc[hh * 4 + 2] * tq[2] + Wc[hh * 4 + 3] * tq[3] + bcv[hh]);
    const float o = w2[0] * hv[0] + w2[1] * hv[1] + w2[2] * hv[2] + w2[3] * hv[3] + b2;

    if (lane == 0) out[img] = fast_sigmoid(o);
}

extern "C" void kernel_launch(void* const* d_in, const int* in_sizes, int n_in,
                              void* d_out, int out_size, void* d_ws, size_t ws_size,
                              hipStream_t stream) {
    const float* x         = (const float*)d_in[0];
    const float* kern      = (const float*)d_in[1];
    const float* cb        = (const float*)d_in[2];
    const float* rot       = (const float*)d_in[3];
    const float* ent       = (const float*)d_in[4];
    const float* fc_map_w  = (const float*)d_in[5];
    const float* fc_map_b  = (const float*)d_in[6];
    const float* qfc_w     = (const float*)d_in[7];
    const float* qfc_b     = (const float*)d_in[8];
    const float* to_qcnn_w = (const float*)d_in[9];
    const float* to_qcnn_b = (const float*)d_in[10];
    const float* qcnn_w1   = (const float*)d_in[11];
    const float* qcnn_b1   = (const float*)d_in[12];
    const float* qcnn_w2   = (const float*)d_in[13];
    const float* qcnn_b2   = (const float*)d_in[14];
    float* out = (float*)d_out;
    float* ws  = (float*)d_ws;

    const int B = in_sizes[0] / IMG_ELEMS;

    setup_compose<<<1, 256, 0, stream>>>(fc_map_w, fc_map_b, qfc_w, qfc_b,
                                         to_qcnn_w, to_qcnn_b, qcnn_w1, qcnn_b1,
                                         qcnn_w2, qcnn_b2, ws);

    const int wavesPerBlock = 8;                       // 256 threads = 8 wave32
    const int grid = (B + wavesPerBlock - 1) / wavesPerBlock;
    hybrid_conv_attn<<<grid, 256, 0, stream>>>(x, kern, cb, rot, ent, ws, out, B);
}